// RGCN_62191126446311
// MI455X (gfx1250) — compile-verified
//
#include <hip/hip_runtime.h>

// CDNA5 / gfx1250 RGCN forward — pull-based (CSR) aggregation + FP32 WMMA GEMMs.
//
// Roofline: 32 GFLOP of GEMM (negligible on WMMA) vs ~3 GB of edge gather traffic,
// almost all L2-resident (node buffers are 51 MB << 192 MB L2). v2 removes the
// ~624M global f32 atomics of the push-scatter design: build CSR once per call
// (counts -> scan -> fill), then each layer aggregates gather-style with zero
// atomics and exactly-once coalesced stores. Last layer projects 128->32 with
// WMMA *before* the edge pass (4x less edge traffic).

typedef __attribute__((ext_vector_type(2))) float v2f;
typedef __attribute__((ext_vector_type(8))) float v8f;

#define NNODE 100000
#define NEDGE 500000
#define NREL  3

// ---------------------------------------------------------------- zero (float4 stores)
__global__ void k_zero(float* __restrict__ p, long n) {
    long i = ((long)blockIdx.x * blockDim.x + threadIdx.x) * 4;
    if (i + 3 < n) *(float4*)(p + i) = make_float4(0.f, 0.f, 0.f, 0.f);
}

// ---------------------------------------------------------------- degree count (f32 atomics, tiny)
__global__ void k_count(const int* __restrict__ idx, float* __restrict__ cnt,
                        int E_, int Nn, long total) {
    long t = (long)blockIdx.x * blockDim.x + threadIdx.x;
    if (t >= total) return;
    int r = (int)(t / E_);
    atomicAdd(&cnt[(long)r * Nn + idx[t]], 1.0f);
}

// cnt = clip(deg,1)^-0.5
__global__ void k_rsqrt(float* __restrict__ p, long n) {
    long t = (long)blockIdx.x * blockDim.x + threadIdx.x;
    if (t >= n) return;
    p[t] = rsqrtf(fmaxf(p[t], 1.0f));
}

// ---------------------------------------------------------------- exclusive scan of in-degrees
// Single workgroup, 1024 threads, LDS Hillis-Steele per chunk with running carry.
// n = 100k -> ~98 chunks; microseconds. Reads float counts (exact ints < 2^24).
__global__ void k_scan(const float* __restrict__ cnt, int* __restrict__ rowptr, int n) {
    __shared__ int buf[1024];
    __shared__ int carry_s;
    if (threadIdx.x == 0) carry_s = 0;
    __syncthreads();
    for (int base = 0; base < n; base += 1024) {
        int i = base + (int)threadIdx.x;
        int v = (i < n) ? (int)cnt[i] : 0;
        int c = carry_s;
        buf[threadIdx.x] = v;
        __syncthreads();
        for (int off = 1; off < 1024; off <<= 1) {
            int t = (threadIdx.x >= off) ? buf[threadIdx.x - off] : 0;
            __syncthreads();
            buf[threadIdx.x] += t;
            __syncthreads();
        }
        if (i < n) rowptr[i] = c + buf[threadIdx.x] - v;  // exclusive
        __syncthreads();
        if (threadIdx.x == 1023) carry_s = c + buf[1023];
        __syncthreads();
    }
    if (threadIdx.x == 0) rowptr[n] = carry_s;
}

// ---------------------------------------------------------------- CSR fill (once per call)
// csr payload = src id + fused edge weight dout[src]*din[dst]
__global__ void k_fill(const int* __restrict__ src, const int* __restrict__ dst,
                       const float* __restrict__ nsrc, const float* __restrict__ ndst,
                       const int* __restrict__ rowptr, int* __restrict__ cursor,
                       int* __restrict__ csr_s, float* __restrict__ csr_w, int E_) {
    int t = blockIdx.x * blockDim.x + threadIdx.x;
    if (t >= E_) return;
    int s = src[t], d = dst[t];
    int pos = rowptr[d] + atomicAdd(&cursor[d], 1);
    csr_s[pos] = s;
    csr_w[pos] = nsrc[s] * ndst[d];
}

// ---------------------------------------------------------------- pull aggregation, 128-wide
// One wave per destination node; lane holds 4 features (float4). Per in-edge:
// one coalesced 512B row gather of h[src] (L2-hot) + register FMA. No atomics.
__global__ void k_gather128(const float* __restrict__ h, const int* __restrict__ csr_s,
                            const float* __restrict__ csr_w, const int* __restrict__ rowptr,
                            float* __restrict__ agg, int Nn) {
    const int wave = threadIdx.x >> 5;
    const int lane = threadIdx.x & 31;
    const int node = blockIdx.x * 8 + wave;
    if (node >= Nn) return;
    const int beg = rowptr[node], end = rowptr[node + 1];
    const int col = lane * 4;
    float a0 = 0.f, a1 = 0.f, a2 = 0.f, a3 = 0.f;
    for (int e = beg; e < end; ++e) {
        int s = csr_s[e];
        float w = csr_w[e];
        float4 v = *(const float4*)(h + (long)s * 128 + col);
        a0 = fmaf(v.x, w, a0);
        a1 = fmaf(v.y, w, a1);
        a2 = fmaf(v.z, w, a2);
        a3 = fmaf(v.w, w, a3);
    }
    *(float4*)(agg + (long)node * 128 + col) = make_float4(a0, a1, a2, a3);
}

// ---------------------------------------------------------------- pull aggregation, 32-wide
// One wave per node, lane = one output channel; ACCUM sums across relations.
template <bool ACCUM>
__global__ void k_gather32(const float* __restrict__ p, const int* __restrict__ csr_s,
                           const float* __restrict__ csr_w, const int* __restrict__ rowptr,
                           float* __restrict__ out, int Nn) {
    const int wave = threadIdx.x >> 5;
    const int lane = threadIdx.x & 31;
    const int node = blockIdx.x * 8 + wave;
    if (node >= Nn) return;
    const int beg = rowptr[node], end = rowptr[node + 1];
    float acc = 0.f;
    for (int e = beg; e < end; ++e)
        acc = fmaf(p[(long)csr_s[e] * 32 + lane], csr_w[e], acc);
    long o = (long)node * 32 + lane;
    if (ACCUM) out[o] += acc;
    else       out[o] = acc;
}

// ---------------------------------------------------------------- FP32 WMMA GEMM
// C[16 x OUTF strip] = A[16 x 128] * W[128 x OUTF] (+bias)(ReLU?)(+=?)
// One wave per 16-row strip; A fragment reused across NT column tiles.
// A 16x4 f32 layout (ISA 7.12.2): lane = M + 16*(K/2), vgpr = K&1.
template <int OUTF, int NT, bool RELU, bool ACCUM, bool BIAS>
__global__ void k_gemm(const float* __restrict__ A, const float* __restrict__ W,
                       const float* __restrict__ bias, float* __restrict__ C, int nTiles) {
    const int wave = threadIdx.x >> 5;  // 4 waves / block
    const int lane = threadIdx.x & 31;
    const int tile = blockIdx.x * 4 + wave;
    if (tile >= nTiles) return;         // wave-uniform: EXEC all-ones for WMMA

    const int m  = lane & 15;
    const int kh = lane >> 4;

    const long row0 = (long)tile * 16;
    const float* Arow = A + (row0 + m) * 128 + 2 * kh;

    v8f acc[NT] = {};

#pragma unroll 4
    for (int k0 = 0; k0 < 128; k0 += 4) {
        v2f a = *(const v2f*)(Arow + k0);
#pragma unroll
        for (int j = 0; j < NT; ++j) {
            int col = j * 16 + m;
            v2f b;
            b.x = W[(k0 + 2 * kh + 0) * OUTF + col];
            b.y = W[(k0 + 2 * kh + 1) * OUTF + col];
            acc[j] = __builtin_amdgcn_wmma_f32_16x16x4_f32(
                false, a, false, b, (short)0, acc[j], false, false);
        }
    }

#pragma unroll
    for (int j = 0; j < NT; ++j) {
        int col = j * 16 + m;
        float bv = BIAS ? bias[col] : 0.f;
#pragma unroll
        for (int i = 0; i < 8; ++i) {
            long row = row0 + i + 8 * kh;  // C layout: vgpr i -> M = i + 8*(lane/16)
            float v = acc[j][i] + bv;
            if (RELU) v = fmaxf(v, 0.f);
            if (ACCUM) C[row * OUTF + col] += v;
            else       C[row * OUTF + col]  = v;
        }
    }
}

// out[n, c] += sum_r b3[r][c]
__global__ void k_bias_out(float* __restrict__ out, const float* __restrict__ b3, long n) {
    long t = (long)blockIdx.x * blockDim.x + threadIdx.x;
    if (t >= n) return;
    int c = (int)(t & 31);
    out[t] += b3[c] + b3[32 + c] + b3[64 + c];
}

// ---------------------------------------------------------------- launch
extern "C" void kernel_launch(void* const* d_in, const int* in_sizes, int n_in,
                              void* d_out, int out_size, void* d_ws, size_t ws_size,
                              hipStream_t stream) {
    const float* x   = (const float*)d_in[0];
    const int*   src = (const int*)d_in[1];
    const int*   dst = (const int*)d_in[2];
    const float* W0  = (const float*)d_in[3];
    const float* b0  = (const float*)d_in[4];
    const float* W1  = (const float*)d_in[5];
    const float* b1  = (const float*)d_in[6];
    const float* W2  = (const float*)d_in[7];
    const float* b2  = (const float*)d_in[8];
    const float* W3  = (const float*)d_in[9];
    const float* b3  = (const float*)d_in[10];
    float* out = (float*)d_out;

    // workspace layout (all 4-byte types)
    float* ws     = (float*)d_ws;
    float* nout   = ws;                                   // [R,N] src norms
    float* nin    = nout + (long)NREL * NNODE;            // [R,N] dst norms
    int*   rowptr = (int*)(nin + (long)NREL * NNODE);     // [R, N+1]
    int*   cursor = rowptr + (long)NREL * (NNODE + 1);    // [R, N]
    int*   csr_s  = cursor + (long)NREL * NNODE;          // [R, E]
    float* csr_w  = (float*)(csr_s + (long)NREL * NEDGE); // [R, E]
    float* bufA   = csr_w + (long)NREL * NEDGE;           // [N,128]
    float* bufB   = bufA + (long)NNODE * 128;             // [N,128]
    float* agg    = bufB + (long)NNODE * 128;             // [N,128]; reused as p[3][N,32]

    const int tiles = NNODE / 16;        // 6250 (exact)
    const int gblk  = (tiles + 3) / 4;   // 4 WMMA waves / block
    const int nblk  = (NNODE + 7) / 8;   // 8 gather waves / block

    // ---- degree counts -> rowptr scan -> norms -> CSR fill (once per call)
    {
        long n = 2L * NREL * NNODE;      // nout & nin contiguous
        k_zero<<<(int)((n / 4 + 255) / 256), 256, 0, stream>>>(nout, n);
        long tot = (long)NREL * NEDGE;
        int  blk = (int)((tot + 255) / 256);
        k_count<<<blk, 256, 0, stream>>>(src, nout, NEDGE, NNODE, tot);
        k_count<<<blk, 256, 0, stream>>>(dst, nin,  NEDGE, NNODE, tot);
        for (int r = 0; r < NREL; ++r)
            k_scan<<<1, 1024, 0, stream>>>(nin + (long)r * NNODE,
                                           rowptr + (long)r * (NNODE + 1), NNODE);
        k_rsqrt<<<(int)((n + 255) / 256), 256, 0, stream>>>(nout, n);
        long nc = (long)NREL * NNODE;
        k_zero<<<(int)((nc / 4 + 255) / 256), 256, 0, stream>>>((float*)cursor, nc);
        int eblk = (NEDGE + 255) / 256;
        for (int r = 0; r < NREL; ++r)
            k_fill<<<eblk, 256, 0, stream>>>(
                src + (long)r * NEDGE, dst + (long)r * NEDGE,
                nout + (long)r * NNODE, nin + (long)r * NNODE,
                rowptr + (long)r * (NNODE + 1), cursor + (long)r * NNODE,
                csr_s + (long)r * NEDGE, csr_w + (long)r * NEDGE, NEDGE);
    }

    // ---- layers 0..2: pull-gather(128) -> WMMA GEMM 128x128 + bias + ReLU, sum over relations
    auto layer = [&](const float* hin, float* hout, const float* W, const float* b) {
        for (int r = 0; r < NREL; ++r) {
            k_gather128<<<nblk, 256, 0, stream>>>(
                hin, csr_s + (long)r * NEDGE, csr_w + (long)r * NEDGE,
                rowptr + (long)r * (NNODE + 1), agg, NNODE);
            if (r == 0)
                k_gemm<128, 8, true, false, true><<<gblk, 128, 0, stream>>>(
                    agg, W + (long)r * 128 * 128, b + r * 128, hout, tiles);
            else
                k_gemm<128, 8, true, true, true><<<gblk, 128, 0, stream>>>(
                    agg, W + (long)r * 128 * 128, b + r * 128, hout, tiles);
        }
    };
    layer(x,    bufA, W0, b0);
    layer(bufA, bufB, W1, b1);
    layer(bufB, bufA, W2, b2);

    // ---- layer 3 (linear): WMMA project 128->32 first, then 32-wide pull-gather
    for (int r = 0; r < NREL; ++r)
        k_gemm<32, 2, false, false, false><<<gblk, 128, 0, stream>>>(
            bufA, W3 + (long)r * 128 * 32, nullptr, agg + (long)r * NNODE * 32, tiles);
    for (int r = 0; r < NREL; ++r) {
        if (r == 0)
            k_gather32<false><<<nblk, 256, 0, stream>>>(
                agg + (long)r * NNODE * 32, csr_s + (long)r * NEDGE, csr_w + (long)r * NEDGE,
                rowptr + (long)r * (NNODE + 1), out, NNODE);
        else
            k_gather32<true><<<nblk, 256, 0, stream>>>(
                agg + (long)r * NNODE * 32, csr_s + (long)r * NEDGE, csr_w + (long)r * NEDGE,
                rowptr + (long)r * (NNODE + 1), out, NNODE);
    }
    {
        long n = (long)NNODE * 32;
        k_bias_out<<<(int)((n + 255) / 256), 256, 0, stream>>>(out, b3, n);
    }
}